// PCCE_PERR_45578192945702
// MI455X (gfx1250) — compile-verified
//
#include <hip/hip_runtime.h>

// CDNA5 / gfx1250. Wave32. WMMA-based wave reduction (V_WMMA_F32_16X16X4_F32).

typedef float v2f __attribute__((ext_vector_type(2)));
typedef float v8f __attribute__((ext_vector_type(8)));

#define BLOCK 256
#define LAMBDA_W 1.5f  // 1 + LAMBDA_0

// LABEL_INFO = [2,1,1,0,0] packed 2 bits per class: 2 | 1<<2 | 1<<4 = 0x16
__device__ __forceinline__ int group_of(int c) { return (0x16 >> (c << 1)) & 3; }

__device__ __forceinline__ float row_term(float x0, float x1, float x2, float x3,
                                          float x4, int label) {
  // argmax with first-max tie semantics (strict >), and running max
  float m = x0; int idx = 0;
  if (x1 > m) { m = x1; idx = 1; }
  if (x2 > m) { m = x2; idx = 2; }
  if (x3 > m) { m = x3; idx = 3; }
  if (x4 > m) { m = x4; idx = 4; }
  float se = __expf(x0 - m) + __expf(x1 - m) + __expf(x2 - m) +
             __expf(x3 - m) + __expf(x4 - m);
  float lse = m + __logf(se);
  // Branchless gather of x[label]: sequential cndmask chain (no branch tree).
  float xl = x0;
  xl = (label == 1) ? x1 : xl;
  xl = (label == 2) ? x2 : xl;
  xl = (label == 3) ? x3 : xl;
  xl = (label == 4) ? x4 : xl;
  float ce = lse - xl;
  float w  = (group_of(label) != group_of(idx)) ? LAMBDA_W : 1.0f;
  return ce * w;
}

// Each thread handles 4 consecutive rows (20 floats = five aligned b128 loads).
// Grid exactly covers T*B rows; no divergence -> EXEC all-ones at the WMMA.
__global__ void __launch_bounds__(BLOCK) pcce_partial_kernel(
    const float* __restrict__ preds, const int* __restrict__ y,
    float* __restrict__ partial, int bMask) {
  const int tid = blockIdx.x * BLOCK + threadIdx.x;
  const long long r0 = (long long)tid * 4;             // first row for this thread
  const float4* p4 = (const float4*)(preds + r0 * 5);  // r0*20 bytes, 16B aligned
  float4 q0 = p4[0];
  float4 q1 = p4[1];
  float4 q2 = p4[2];
  float4 q3 = p4[3];
  float4 q4 = p4[4];

  const int b0 = (int)(r0 & (long long)bMask);         // b = r mod B; 4 consecutive
  const int4 lab = *(const int4*)(y + b0);             // 16B aligned (b0 % 4 == 0)

  float acc0 = row_term(q0.x, q0.y, q0.z, q0.w, q1.x, lab.x) +
               row_term(q1.y, q1.z, q1.w, q2.x, q2.y, lab.y);
  float acc1 = row_term(q2.z, q2.w, q3.x, q3.y, q3.z, lab.z) +
               row_term(q3.w, q4.x, q4.y, q4.z, q4.w, lab.w);

  // ---- wave32 reduction via V_WMMA_F32_16X16X4_F32 ----
  // A (16x4 f32, 2 VGPRs/lane) carries all 64 per-wave partials; B = ones.
  // D[m][n] = sum_k A[m][k]; columns are replicated across lanes 0..15 / 16..31.
  v2f a;  a.x = acc0; a.y = acc1;
  v2f bv; bv.x = 1.0f; bv.y = 1.0f;
  v8f c = {};
  v8f d = __builtin_amdgcn_wmma_f32_16x16x4_f32(
      /*neg_a=*/false, a, /*neg_b=*/false, bv,
      /*c_mod=*/(short)0, c, /*reuse_a=*/false, /*reuse_b=*/false);
  float s = ((d[0] + d[1]) + (d[2] + d[3])) + ((d[4] + d[5]) + (d[6] + d[7]));
  s += __shfl_xor(s, 16, 32);  // combine M=0..7 half with M=8..15 half -> wave sum

  __shared__ float lds[BLOCK / 32];
  const int lane = threadIdx.x & 31;
  const int wave = threadIdx.x >> 5;
  if (lane == 0) lds[wave] = s;
  __syncthreads();
  if (threadIdx.x == 0) {
    float bs = 0.0f;
#pragma unroll
    for (int i = 0; i < BLOCK / 32; ++i) bs += lds[i];
    partial[blockIdx.x] = bs;
  }
}

// Deterministic single-block tree reduction of per-block partials.
__global__ void __launch_bounds__(BLOCK) pcce_final_kernel(
    const float* __restrict__ partial, float* __restrict__ out, int n,
    float invB) {
  float s = 0.0f;
  for (int i = threadIdx.x; i < n; i += BLOCK) s += partial[i];
#pragma unroll
  for (int off = 16; off >= 1; off >>= 1) s += __shfl_xor(s, off, 32);
  __shared__ float lds[BLOCK / 32];
  const int lane = threadIdx.x & 31;
  const int wave = threadIdx.x >> 5;
  if (lane == 0) lds[wave] = s;
  __syncthreads();
  if (threadIdx.x == 0) {
    float t = 0.0f;
#pragma unroll
    for (int i = 0; i < BLOCK / 32; ++i) t += lds[i];
    out[0] = t * invB;  // sum_t mean_b = total / B
  }
}

extern "C" void kernel_launch(void* const* d_in, const int* in_sizes, int n_in,
                              void* d_out, int out_size, void* d_ws,
                              size_t ws_size, hipStream_t stream) {
  const float* preds = (const float*)d_in[0];  // [T, B, C] f32
  const int* y = (const int*)d_in[1];          // [B] int32
  const long long totalRows = (long long)in_sizes[0] / 5;  // T*B = 4,194,304
  const int B = in_sizes[1];                               // 1,048,576 (pow2)
  const int threads = (int)(totalRows / 4);                // 1,048,576
  const int blocks = threads / BLOCK;                      // 4096

  float* partial = (float*)d_ws;  // 4096 floats, fully overwritten each call

  pcce_partial_kernel<<<blocks, BLOCK, 0, stream>>>(preds, y, partial, B - 1);
  pcce_final_kernel<<<1, BLOCK, 0, stream>>>(partial, (float*)d_out, blocks,
                                             1.0f / (float)B);
}